// PrimalDualNetwork_2035814499124
// MI455X (gfx1250) — compile-verified
//
#include <hip/hip_runtime.h>
#include <cstdint>
#include <cstddef>

// Problem constants (fixed by the reference).
#define H_DIM 4096
#define W_DIM 4096
#define TILE  1024           // pixels per block (256 threads x float4)
#define MAX_IT 20

static constexpr float SIGMA_C = 1.0f / (7.0f * 0.01f);
static constexpr float TAU_C   = 0.01f;
static constexpr float THETA_C = 0.5f;
static constexpr float LT_C    = 7.0f * 0.01f;          // lambda * tau
static constexpr float INV_C   = 1.0f / (1.0f + LT_C);  // 1/(1+lambda*tau)

typedef __attribute__((ext_vector_type(4))) float v4f;

// ---------------------------------------------------------------------------
// CDNA5 async global -> LDS staging (tracked with ASYNCcnt).
// GV addressing mode: per-lane 64-bit global address, per-lane LDS byte addr.
// ---------------------------------------------------------------------------
__device__ __forceinline__ uint32_t lds_addr(const void* p) {
  // Generic LDS pointers carry the raw DS offset in the low 32 bits
  // (aperture in the high bits) -- truncation yields the DS address.
  return (uint32_t)(uintptr_t)p;
}

__device__ __forceinline__ void async_ld_b128(uint32_t lds_off, const void* gptr) {
  uint64_t ga = (uint64_t)(uintptr_t)gptr;
  asm volatile("global_load_async_to_lds_b128 %0, %1, off"
               :: "v"(lds_off), "v"(ga)
               : "memory");
}

__device__ __forceinline__ void wait_async0() {
#if __has_builtin(__builtin_amdgcn_s_wait_asynccnt)
  __builtin_amdgcn_s_wait_asynccnt(0);
#else
  asm volatile("s_wait_asynccnt 0" ::: "memory");
#endif
}

__device__ __forceinline__ float clamp1(float v) {
  return fminf(fmaxf(v, -1.0f), 1.0f);
}

// ---------------------------------------------------------------------------
// Init: x = img, x_tilde = img, y = 0.   (float4 streams)
// ---------------------------------------------------------------------------
__global__ __launch_bounds__(256) void pd_init(const float* __restrict__ img,
                                               float* __restrict__ x,
                                               float* __restrict__ xt,
                                               float* __restrict__ y0,
                                               float* __restrict__ y1) {
  size_t idx = ((size_t)blockIdx.x * blockDim.x + threadIdx.x) * 4;
  v4f v = *reinterpret_cast<const v4f*>(img + idx);
  *reinterpret_cast<v4f*>(x  + idx) = v;
  *reinterpret_cast<v4f*>(xt + idx) = v;
  v4f z = (v4f){0.f, 0.f, 0.f, 0.f};
  *reinterpret_cast<v4f*>(y0 + idx) = z;
  *reinterpret_cast<v4f*>(y1 + idx) = z;
}

// ---------------------------------------------------------------------------
// Dual update:  y = clamp(y + SIGMA * fwd_grad(x_tilde) * w, -1, 1)
//   gh[i][j] = (j<W-1) ? xt[i][j+1]-xt[i][j] : 0
//   gv[i][j] = (i<H-1) ? xt[i+1][j]-xt[i][j] : 0
// x_tilde rows staged into LDS via async b128 (current row gets +4 halo).
// w is read-once per iteration -> nontemporal (protects y/x_tilde in L2).
// ---------------------------------------------------------------------------
__global__ __launch_bounds__(256) void pd_dual(const float* __restrict__ xt,
                                               const float* __restrict__ w0,
                                               const float* __restrict__ w1,
                                               float* __restrict__ y0,
                                               float* __restrict__ y1) {
  __shared__ float s_c[TILE + 4];   // current row + right halo
  __shared__ float s_d[TILE];       // row below

  const int t  = threadIdx.x;
  const int i  = blockIdx.y;
  const int j0 = blockIdx.x * TILE;
  const int id = (i < H_DIM - 1) ? (i + 1) : i;   // clamped; gv predicated anyway

  const float* rowc = xt + (size_t)i  * W_DIM + j0;
  const float* rowd = xt + (size_t)id * W_DIM + j0;

  async_ld_b128(lds_addr(&s_c[t * 4]), rowc + t * 4);
  async_ld_b128(lds_addr(&s_d[t * 4]), rowd + t * 4);
  if (t == 0 && (j0 + TILE) < W_DIM) {
    async_ld_b128(lds_addr(&s_c[TILE]), rowc + TILE);   // right halo (16B, in-row)
  }
  wait_async0();
  __syncthreads();

  const size_t base = (size_t)i * W_DIM + j0 + t * 4;
  v4f yv0 = *reinterpret_cast<const v4f*>(y0 + base);
  v4f yv1 = *reinterpret_cast<const v4f*>(y1 + base);
  v4f wv0 = __builtin_nontemporal_load(reinterpret_cast<const v4f*>(w0 + base));
  v4f wv1 = __builtin_nontemporal_load(reinterpret_cast<const v4f*>(w1 + base));

  v4f c = *reinterpret_cast<const v4f*>(&s_c[t * 4]);
  const float cnext = s_c[t * 4 + 4];               // cross-thread / halo element
  v4f d = *reinterpret_cast<const v4f*>(&s_d[t * 4]);

  v4f r0, r1;
#pragma unroll
  for (int k = 0; k < 4; ++k) {
    const int gj = j0 + t * 4 + k;
    const float xr = (k < 3) ? c[k + 1] : cnext;    // resolved at unroll time
    const float gh = (gj < W_DIM - 1) ? (xr   - c[k]) : 0.0f;
    const float gv = (i  < H_DIM - 1) ? (d[k] - c[k]) : 0.0f;
    r0[k] = clamp1(yv0[k] + SIGMA_C * gh * wv0[k]);
    r1[k] = clamp1(yv1[k] + SIGMA_C * gv * wv1[k]);
  }

  *reinterpret_cast<v4f*>(y0 + base) = r0;
  *reinterpret_cast<v4f*>(y1 + base) = r1;
}

// ---------------------------------------------------------------------------
// Primal update:
//   dh[i][j] = (j<W-1 ? y0[i][j] : 0) - (j>0 ? y0[i][j-1] : 0)
//   dv[i][j] = (i<H-1 ? y1[i][j] : 0) - (i>0 ? y1[i-1][j] : 0)
//   x_new    = (x + TAU*(dh+dv) + LT*img) / (1+LT)
//   x_tilde  = x_new + THETA*(x_new - x_old)
// y rows staged into LDS via async b128 (y0 row gets a left halo).
// img is read-once per iteration -> nontemporal.
// ---------------------------------------------------------------------------
__global__ __launch_bounds__(256) void pd_primal(const float* __restrict__ y0,
                                                 const float* __restrict__ y1,
                                                 const float* __restrict__ img,
                                                 float* __restrict__ x,
                                                 float* __restrict__ xt_out) {
  __shared__ float s_h[TILE + 4];   // y0 row, left halo in [0..3]
  __shared__ float s_vc[TILE];      // y1 row i
  __shared__ float s_vu[TILE];      // y1 row i-1

  const int t  = threadIdx.x;
  const int i  = blockIdx.y;
  const int j0 = blockIdx.x * TILE;
  const int iu = (i > 0) ? (i - 1) : 0;           // clamped; dv predicated anyway

  const float* ry0 = y0 + (size_t)i  * W_DIM + j0;
  const float* ryc = y1 + (size_t)i  * W_DIM + j0;
  const float* ryu = y1 + (size_t)iu * W_DIM + j0;

  async_ld_b128(lds_addr(&s_h[4 + t * 4]), ry0 + t * 4);
  async_ld_b128(lds_addr(&s_vc[t * 4]),    ryc + t * 4);
  async_ld_b128(lds_addr(&s_vu[t * 4]),    ryu + t * 4);
  if (t == 0 && j0 > 0) {
    async_ld_b128(lds_addr(&s_h[0]), ry0 - 4);    // left halo (16B, in-row)
  }
  wait_async0();
  __syncthreads();

  const size_t base = (size_t)i * W_DIM + j0 + t * 4;
  v4f xv = *reinterpret_cast<const v4f*>(x + base);
  v4f gv = __builtin_nontemporal_load(reinterpret_cast<const v4f*>(img + base));

  v4f h = *reinterpret_cast<const v4f*>(&s_h[4 + t * 4]);
  const float hprev = s_h[3 + t * 4];             // cross-thread / halo element
  v4f vc = *reinterpret_cast<const v4f*>(&s_vc[t * 4]);
  v4f vu = *reinterpret_cast<const v4f*>(&s_vu[t * 4]);

  v4f xn, xe;
#pragma unroll
  for (int k = 0; k < 4; ++k) {
    const int gj = j0 + t * 4 + k;
    const float hl = (k > 0) ? h[k - 1] : hprev;  // resolved at unroll time
    const float dh = ((gj < W_DIM - 1) ? h[k]  : 0.0f)
                   - ((gj > 0)         ? hl    : 0.0f);
    const float dv = ((i  < H_DIM - 1) ? vc[k] : 0.0f)
                   - ((i  > 0)         ? vu[k] : 0.0f);
    const float xnew = (xv[k] + TAU_C * (dh + dv) + LT_C * gv[k]) * INV_C;
    xn[k] = xnew;
    xe[k] = xnew + THETA_C * (xnew - xv[k]);
  }

  *reinterpret_cast<v4f*>(x      + base) = xn;
  *reinterpret_cast<v4f*>(xt_out + base) = xe;
}

// ---------------------------------------------------------------------------
// Host launcher.
//   d_in[0] = img_obs (1,H,W) f32        d_in[1] = w (2,H,W) f32
//   d_out   = x_tilde (1,H,W) f32
//   d_ws    : x (N), y0 (N), y1 (N)  -> 3*N*4 bytes
// ---------------------------------------------------------------------------
extern "C" void kernel_launch(void* const* d_in, const int* in_sizes, int n_in,
                              void* d_out, int out_size, void* d_ws, size_t ws_size,
                              hipStream_t stream) {
  (void)in_sizes; (void)n_in; (void)out_size; (void)ws_size;

  const size_t N = (size_t)H_DIM * W_DIM;

  const float* img = (const float*)d_in[0];
  const float* w0  = (const float*)d_in[1];        // w[0]
  const float* w1  = (const float*)d_in[1] + N;    // w[1]

  float* xt = (float*)d_out;                       // x_tilde lives in d_out
  float* ws = (float*)d_ws;
  float* x  = ws;                                  // primal iterate
  float* y0 = ws + N;                              // dual, horizontal
  float* y1 = ws + 2 * N;                          // dual, vertical

  const int initBlocks = (int)(N / (4 * 256));     // float4 per thread
  pd_init<<<initBlocks, 256, 0, stream>>>(img, x, xt, y0, y1);

  dim3 grid(W_DIM / TILE, H_DIM);
  for (int it = 0; it < MAX_IT; ++it) {
    pd_dual<<<grid, 256, 0, stream>>>(xt, w0, w1, y0, y1);
    pd_primal<<<grid, 256, 0, stream>>>(y0, y1, img, x, xt);
  }
}